// ConvolutionalLUTLayer_22101901705996
// MI455X (gfx1250) — compile-verified
//
#include <hip/hip_runtime.h>
#include <hip/hip_bf16.h>

// DWN convolutional LUT layer for MI455X (gfx1250).
//   x:    [32,1,32,32] f32      lut0: [64,16,16] f32   lut1: [64,4,16] f32   lut2: [64,1,16] f32
//   idx0: [64,16,4] i32 (in [0,25))   idx1: [64,4,4] i32 (in [0,16))   idx2: [64,1,4] i32 (in [0,4))
//   out:  [32,64,28,28] f32
//
// One wave per output channel; one block = 8 channels x 1 image; 2 pixels per lane per
// iteration (p and p+392; since 392 = 14*28 the second pixel is exactly 14 rows below,
// i.e. a constant +448 LDS offset -> one address per gather + immediate offsets).
// Image DMA'd to LDS with the gfx1250 async-to-LDS path. LUTs are stored in LDS
// pre-sigmoided AND pre-differenced as (base, delta) pairs so the first contraction
// level is a pure FMA (22 VALU/node instead of 30).

#define PIX   784   // 28*28
#define HALF  392   // PIX/2 = 14 rows of 28
#define BOFF  448   // 14 rows * 32 floats: LDS offset between pixel A and pixel B

// Multilinear interpolation of a 16-entry LUT for TWO pixels sharing one table fetch.
// Table layout per node: pair q (q = s0*4+s1*2+s2) at [2q] = base e_q = sig(w[2q]),
// [2q+1] = delta d_q = sig(w[2q+1]) - sig(w[2q]).  Address bits: a = s0*8+s1*4+s2*2+s3.
__device__ __forceinline__ void lut4_eval2(const float* __restrict__ w,
                                           float a0, float a1, float a2, float a3,
                                           float b0, float b1, float b2, float b3,
                                           float& rA, float& rB) {
    float4 w0 = *(const float4*)(w + 0);   // e0 d0 e1 d1
    float4 w1 = *(const float4*)(w + 4);   // e2 d2 e3 d3
    float4 w2 = *(const float4*)(w + 8);   // e4 d4 e5 d5
    float4 w3 = *(const float4*)(w + 12);  // e6 d6 e7 d7
    // level 1: contract s3 (LSB) -- pure FMA thanks to pre-differenced tables
    float uA0 = fmaf(a3, w0.y, w0.x), uB0 = fmaf(b3, w0.y, w0.x);
    float uA1 = fmaf(a3, w0.w, w0.z), uB1 = fmaf(b3, w0.w, w0.z);
    float uA2 = fmaf(a3, w1.y, w1.x), uB2 = fmaf(b3, w1.y, w1.x);
    float uA3 = fmaf(a3, w1.w, w1.z), uB3 = fmaf(b3, w1.w, w1.z);
    float uA4 = fmaf(a3, w2.y, w2.x), uB4 = fmaf(b3, w2.y, w2.x);
    float uA5 = fmaf(a3, w2.w, w2.z), uB5 = fmaf(b3, w2.w, w2.z);
    float uA6 = fmaf(a3, w3.y, w3.x), uB6 = fmaf(b3, w3.y, w3.x);
    float uA7 = fmaf(a3, w3.w, w3.z), uB7 = fmaf(b3, w3.w, w3.z);
    // level 2: contract s2
    float vA0 = fmaf(a2, uA1 - uA0, uA0), vB0 = fmaf(b2, uB1 - uB0, uB0);
    float vA1 = fmaf(a2, uA3 - uA2, uA2), vB1 = fmaf(b2, uB3 - uB2, uB2);
    float vA2 = fmaf(a2, uA5 - uA4, uA4), vB2 = fmaf(b2, uB5 - uB4, uB4);
    float vA3 = fmaf(a2, uA7 - uA6, uA6), vB3 = fmaf(b2, uB7 - uB6, uB6);
    // level 3: contract s1
    float qA0 = fmaf(a1, vA1 - vA0, vA0), qB0 = fmaf(b1, vB1 - vB0, vB0);
    float qA1 = fmaf(a1, vA3 - vA2, vA2), qB1 = fmaf(b1, vB3 - vB2, vB2);
    // level 4: contract s0 (MSB)
    rA = fmaf(a0, qA1 - qA0, qA0);
    rB = fmaf(b0, qB1 - qB0, qB0);
}

__device__ __forceinline__ float pick4(float a, float b, float c, float d, int k) {
    float r = a;
    r = (k == 1) ? b : r;
    r = (k == 2) ? c : r;
    r = (k == 3) ? d : r;
    return r;
}

__global__ __launch_bounds__(256)
void dwn_conv_lut_kernel(const float* __restrict__ x,
                         const float* __restrict__ lut0,
                         const float* __restrict__ lut1,
                         const float* __restrict__ lut2,
                         const int*   __restrict__ idx0,
                         const int*   __restrict__ idx1,
                         const int*   __restrict__ idx2,
                         float*       __restrict__ out)
{
    __shared__ float imgLDS[1024];        // one 32x32 image, 4 KB
    __shared__ float tlut[8 * 336];       // (base,delta) sigmoid tables per channel
    __shared__ float h0scr[2 * 16 * 256]; // layer-0 outputs, 2 pixels/lane, bank-aligned

    const int tid = threadIdx.x;
    const int cg  = blockIdx.x;           // channel group 0..7
    const int b   = blockIdx.y;           // image 0..31
    const int c0  = cg * 8;

    // ---- Phase 1: async DMA the whole image into LDS (gfx1250 async-tensor path).
    // 256 threads x 16 B = 4096 B, exactly one image.
    {
        unsigned lds = (unsigned)(size_t)(&imgLDS[0]) + (unsigned)tid * 16u;
        const float* g = x + (size_t)b * 1024 + tid * 4;
        asm volatile("global_load_async_to_lds_b128 %0, %1, off"
                     :: "v"(lds), "v"(g) : "memory");
    }

    // ---- Phase 2 (overlaps the DMA): build (base,delta) sigmoid tables.
    // Per channel ch: nodes 0..15 = lut0, 16..19 = lut1, 20 = lut2; 8 pairs per node.
    // Pair j of node nd: tlut[ch*336 + nd*16 + 2j]   = sig(raw[2j])
    //                    tlut[ch*336 + nd*16 + 2j+1] = sig(raw[2j+1]) - sig(raw[2j])
    for (int e = tid; e < 8 * 168; e += 256) {       // 168 pairs per channel
        int ch = e / 168;
        int r  = e - ch * 168;                        // pair within channel
        int nd = r >> 3;                              // node 0..20
        int j  = r & 7;                               // pair 0..7
        int c  = c0 + ch;
        const float* src;
        if (nd < 16)      src = lut0 + c * 256 + nd * 16;
        else if (nd < 20) src = lut1 + c * 64 + (nd - 16) * 16;
        else              src = lut2 + c * 16;
        float r0 = src[2 * j], r1 = src[2 * j + 1];
        float s0 = 1.0f / (1.0f + __expf(-r0));
        float s1 = 1.0f / (1.0f + __expf(-r1));
        int pos = ch * 336 + nd * 16 + 2 * j;
        tlut[pos]     = s0;
        tlut[pos + 1] = s1 - s0;
    }

    // ---- Phase 2b: wave-uniform index tables via scalar loads, packed into SGPRs.
    const int wave = tid >> 5;            // warpSize == 32 on gfx1250
    const int lane = tid & 31;
    const int cs   = __builtin_amdgcn_readfirstlane(c0 + wave);

    __builtin_prefetch(idx0 + cs * 64, 0, 0);  // global_prefetch_b8
    __builtin_prefetch(idx1 + cs * 16, 0, 0);

    unsigned off0[16];                    // 4 packed byte offsets into the 32x32 image
    #pragma unroll
    for (int m = 0; m < 16; ++m) {
        const int* q = idx0 + cs * 64 + m * 4;
        int k0 = q[0], k1 = q[1], k2 = q[2], k3 = q[3];            // in [0,25)
        unsigned o0 = (unsigned)((k0 / 5) * 32 + (k0 % 5));        // <= 132
        unsigned o1 = (unsigned)((k1 / 5) * 32 + (k1 % 5));
        unsigned o2 = (unsigned)((k2 / 5) * 32 + (k2 % 5));
        unsigned o3 = (unsigned)((k3 / 5) * 32 + (k3 % 5));
        off0[m] = o0 | (o1 << 8) | (o2 << 16) | (o3 << 24);
    }
    unsigned off1[4];                     // 4 packed h0-slot indices, each in [0,16)
    #pragma unroll
    for (int n = 0; n < 4; ++n) {
        const int* q = idx1 + cs * 16 + n * 4;
        off1[n] = (unsigned)q[0] | ((unsigned)q[1] << 8)
                | ((unsigned)q[2] << 16) | ((unsigned)q[3] << 24);
    }
    const int* q2 = idx2 + cs * 4;
    int e2x = q2[0], e2y = q2[1], e2z = q2[2], e2w = q2[3];        // in [0,4)

    asm volatile("s_wait_asynccnt 0x0" ::: "memory");
    __syncthreads();

    // ---- Phase 3: wave-per-channel evaluation, 2 pixels (p, p+392) per lane per iter.
    // Pixel B lives exactly 14 image rows below pixel A: baseB = baseA + 448.
    const float* wl0 = &tlut[wave * 336];
    const float* wl1 = wl0 + 256;
    const float* wl2 = wl0 + 320;
    const float* imgA = &imgLDS[0];
    const float* imgB = &imgLDS[BOFF];    // +1792 B: folds into ds_load immediate offsets
    float* slotA = &h0scr[tid];           // node m at slotA[m*256]
    float* slotB = &h0scr[4096 + tid];
    float* outc  = out + (size_t)(b * 64 + (c0 + wave)) * PIX;

    for (int pA = lane; pA < HALF; pA += 32) {
        const int iA   = pA / 28;
        const int jA   = pA - iA * 28;
        const int base = iA * 32 + jA;    // single per-iteration address; B side is +448

        // Layer 0: 16 nodes; table fetched once, applied to both pixels.
        #pragma unroll
        for (int m = 0; m < 16; ++m) {
            const unsigned pk = off0[m];
            const int o0 = base + (int)( pk        & 255u);
            const int o1 = base + (int)((pk >>  8) & 255u);
            const int o2 = base + (int)((pk >> 16) & 255u);
            const int o3 = base + (int)( pk >> 24        );
            float aw, bw;
            lut4_eval2(wl0 + m * 16,
                       imgA[o0], imgA[o1], imgA[o2], imgA[o3],
                       imgB[o0], imgB[o1], imgB[o2], imgB[o3],
                       aw, bw);
            slotA[m * 256] = aw;
            slotB[m * 256] = bw;
        }

        // Layer 1: 4 nodes gathering from the 16 layer-0 values (wave-uniform indices,
        // lane-contiguous LDS scratch -> conflict-free; B side is a +16 KB imm offset).
        float hA0, hA1, hA2, hA3, hB0, hB1, hB2, hB3;
        {
            const unsigned pk = off1[0];
            const int i0 = (int)(pk & 255u) * 256, i1 = (int)((pk >> 8) & 255u) * 256;
            const int i2 = (int)((pk >> 16) & 255u) * 256, i3 = (int)(pk >> 24) * 256;
            lut4_eval2(wl1 + 0,  slotA[i0], slotA[i1], slotA[i2], slotA[i3],
                                 slotB[i0], slotB[i1], slotB[i2], slotB[i3], hA0, hB0);
        }
        {
            const unsigned pk = off1[1];
            const int i0 = (int)(pk & 255u) * 256, i1 = (int)((pk >> 8) & 255u) * 256;
            const int i2 = (int)((pk >> 16) & 255u) * 256, i3 = (int)(pk >> 24) * 256;
            lut4_eval2(wl1 + 16, slotA[i0], slotA[i1], slotA[i2], slotA[i3],
                                 slotB[i0], slotB[i1], slotB[i2], slotB[i3], hA1, hB1);
        }
        {
            const unsigned pk = off1[2];
            const int i0 = (int)(pk & 255u) * 256, i1 = (int)((pk >> 8) & 255u) * 256;
            const int i2 = (int)((pk >> 16) & 255u) * 256, i3 = (int)(pk >> 24) * 256;
            lut4_eval2(wl1 + 32, slotA[i0], slotA[i1], slotA[i2], slotA[i3],
                                 slotB[i0], slotB[i1], slotB[i2], slotB[i3], hA2, hB2);
        }
        {
            const unsigned pk = off1[3];
            const int i0 = (int)(pk & 255u) * 256, i1 = (int)((pk >> 8) & 255u) * 256;
            const int i2 = (int)((pk >> 16) & 255u) * 256, i3 = (int)(pk >> 24) * 256;
            lut4_eval2(wl1 + 48, slotA[i0], slotA[i1], slotA[i2], slotA[i3],
                                 slotB[i0], slotB[i1], slotB[i2], slotB[i3], hA3, hB3);
        }

        // Layer 2: 1 node over the 4 layer-1 values (register selects, uniform index).
        float sA0 = pick4(hA0, hA1, hA2, hA3, e2x), sB0 = pick4(hB0, hB1, hB2, hB3, e2x);
        float sA1 = pick4(hA0, hA1, hA2, hA3, e2y), sB1 = pick4(hB0, hB1, hB2, hB3, e2y);
        float sA2 = pick4(hA0, hA1, hA2, hA3, e2z), sB2 = pick4(hB0, hB1, hB2, hB3, e2z);
        float sA3 = pick4(hA0, hA1, hA2, hA3, e2w), sB3 = pick4(hB0, hB1, hB2, hB3, e2w);

        float rA, rB;
        lut4_eval2(wl2, sA0, sA1, sA2, sA3, sB0, sB1, sB2, sB3, rA, rB);
        outc[pA]        = rA;
        outc[pA + HALF] = rB;
    }
}

extern "C" void kernel_launch(void* const* d_in, const int* in_sizes, int n_in,
                              void* d_out, int out_size, void* d_ws, size_t ws_size,
                              hipStream_t stream) {
    const float* x    = (const float*)d_in[0];
    const float* lut0 = (const float*)d_in[1];
    const float* lut1 = (const float*)d_in[2];
    const float* lut2 = (const float*)d_in[3];
    const int*   idx0 = (const int*)d_in[4];
    const int*   idx1 = (const int*)d_in[5];
    const int*   idx2 = (const int*)d_in[6];
    float* out = (float*)d_out;

    dim3 grid(8, 32);   // 8 channel groups x 32 images
    dim3 block(256);    // 8 waves: one wave per channel
    dwn_conv_lut_kernel<<<grid, block, 0, stream>>>(x, lut0, lut1, lut2,
                                                    idx0, idx1, idx2, out);
}